// TopK_19284403159148
// MI455X (gfx1250) — compile-verified
//
#include <hip/hip_runtime.h>
#include <stdint.h>

#define ROW_LEN 16384
#define BLOCK   512
#define NWAVES  (BLOCK / 32)        // 16 wave32 per block
#define CHUNK   (ROW_LEN / BLOCK)   // 32 elements per thread
#define F4_PER_THREAD (ROW_LEN / (BLOCK * 4))  // 8 float4 per thread

// float -> uint key, monotone with float ordering (descending float == descending key)
__device__ __forceinline__ unsigned f2key(unsigned u) {
  return u ^ ((unsigned)((int)u >> 31) | 0x80000000u);
}

__global__ __launch_bounds__(BLOCK)
void topk_relu_scatter_kernel(const float* __restrict__ x,
                              const int* __restrict__ kptr,
                              float* __restrict__ out) {
  __shared__ float    row[ROW_LEN];            // 64 KB: the row, LDS-resident
  __shared__ unsigned whist[NWAVES][256];      // 16 KB: wave-private histograms
  __shared__ unsigned cnt[BLOCK];              //  2 KB: rank-pass prefix sums
  __shared__ unsigned sb[3];                   // broadcast: prefix, remaining, m

  const int tid  = threadIdx.x;
  const int wave = tid >> 5;
  const float* xrow = x   + (size_t)blockIdx.x * ROW_LEN;
  float*       orow = out + (size_t)blockIdx.x * ROW_LEN;

  // ---- Stage 1: async DMA row -> LDS (CDNA5 async global->LDS path) ----
  #pragma unroll
  for (int j = 0; j < F4_PER_THREAD; ++j) {
    const int f4 = tid + j * BLOCK;
    uint32_t ldsAddr = (uint32_t)(uintptr_t)(&row[f4 * 4]); // low 32b of flat addr == LDS offset
    const float* gp = xrow + f4 * 4;
    asm volatile("global_load_async_to_lds_b128 %0, %1, off"
                 :: "v"(ldsAddr), "v"(gp) : "memory");
  }
  // overlap: zero first-pass histograms while the DMA is in flight
  for (int i = tid; i < NWAVES * 256; i += BLOCK) ((unsigned*)whist)[i] = 0u;
  asm volatile("s_wait_asynccnt 0" ::: "memory");
  __syncthreads();

  // ---- Stage 2: MSB-first radix select of the k-th largest key ----
  unsigned prefix    = 0;
  unsigned remaining = (unsigned)(*kptr);   // k (scalar broadcast load)
  unsigned mcount    = 0;

  #pragma unroll
  for (int b = 24; b >= 0; b -= 8) {
    if (b != 24) {  // first pass was zeroed during DMA
      for (int i = tid; i < NWAVES * 256; i += BLOCK) ((unsigned*)whist)[i] = 0u;
      __syncthreads();
    }

    #pragma unroll
    for (int j = 0; j < CHUNK; ++j) {
      const int i = tid + j * BLOCK;
      const unsigned key = f2key(__float_as_uint(row[i]));
      const bool match = (b == 24) ? true : (((key ^ prefix) >> (b + 8)) == 0u);
      if (match) atomicAdd(&whist[wave][(key >> b) & 0xFFu], 1u);
    }
    __syncthreads();

    // reduce wave-private histograms into whist[0][*]
    if (tid < 256) {
      unsigned s = 0;
      #pragma unroll
      for (int w = 0; w < NWAVES; ++w) s += whist[w][tid];
      whist[0][tid] = s;
    }
    __syncthreads();

    // wave0-parallel descending cumulative scan: lane owns 8 contiguous bins
    if (tid < 32) {
      const int lane = tid;
      unsigned h[8];
      unsigned total = 0;
      #pragma unroll
      for (int i = 0; i < 8; ++i) { h[i] = whist[0][lane * 8 + i]; total += h[i]; }
      // inclusive suffix-scan of per-lane totals across the wave
      unsigned t = total;
      #pragma unroll
      for (int off = 1; off < 32; off <<= 1) {
        const unsigned o = __shfl_down(t, off);
        if (lane + off < 32) t += o;
      }
      unsigned running = t - total;   // count of keys in bins >= (lane+1)*8
      #pragma unroll
      for (int i = 7; i >= 0; --i) {  // walk own bins from high to low
        const unsigned prev = running;
        running += h[i];
        if (running >= remaining && prev < remaining) {  // unique (lane,i)
          sb[0] = prefix | ((unsigned)(lane * 8 + i) << b);
          sb[1] = remaining - prev;   // equals still needed
          sb[2] = h[i];               // elements matching full prefix
        }
      }
    }
    __syncthreads();
    prefix = sb[0]; remaining = sb[1]; mcount = sb[2];
  }

  const unsigned Tkey = prefix;     // key of the k-th largest element
  const unsigned e    = remaining;  // how many elements equal to T are selected
  // relu(decode(Tkey)): key MSB set => original x >= 0
  const float reluT = (Tkey & 0x80000000u) ? __uint_as_float(Tkey & 0x7FFFFFFFu) : 0.0f;

  // ---- Stage 3 (rare): exact tie-break — lowest indices among equals win ----
  if (mcount > e) {
    const int base = tid * CHUNK;   // contiguous chunks => index-ordered ranks
    unsigned c = 0;
    #pragma unroll
    for (int j = 0; j < CHUNK; ++j)
      c += (f2key(__float_as_uint(row[base + j])) == Tkey);
    cnt[tid] = c;
    __syncthreads();
    if (tid == 0) {                 // serial exclusive scan (rare path)
      unsigned run = 0;
      for (int t = 0; t < BLOCK; ++t) { const unsigned cc = cnt[t]; cnt[t] = run; run += cc; }
    }
    __syncthreads();
    unsigned rank = cnt[tid];
    #pragma unroll
    for (int j = 0; j < CHUNK; ++j) {
      if (f2key(__float_as_uint(row[base + j])) == Tkey) {
        if (rank < e) row[base + j] = __uint_as_float(0x7F800000u); // +inf sentinel = selected
        ++rank;
      }
    }
    __syncthreads();
  }

  // ---- Stage 4: single coalesced float4 output pass ----
  const bool allEq = (mcount == e);
  #pragma unroll
  for (int j = 0; j < F4_PER_THREAD; ++j) {
    const int f4 = tid + j * BLOCK;
    const float4 v = reinterpret_cast<const float4*>(row)[f4];  // ds_load_b128
    const float in[4] = {v.x, v.y, v.z, v.w};
    float o[4];
    #pragma unroll
    for (int c = 0; c < 4; ++c) {
      const unsigned u   = __float_as_uint(in[c]);
      const unsigned key = f2key(u);
      float ov;
      if (key > Tkey)       ov = (u == 0x7F800000u) ? reluT : fmaxf(in[c], 0.0f);
      else if (key == Tkey) ov = allEq ? reluT : 0.0f;
      else                  ov = 0.0f;
      o[c] = ov;
    }
    reinterpret_cast<float4*>(orow)[f4] = make_float4(o[0], o[1], o[2], o[3]);
  }
}

extern "C" void kernel_launch(void* const* d_in, const int* in_sizes, int n_in,
                              void* d_out, int out_size, void* d_ws, size_t ws_size,
                              hipStream_t stream) {
  const float* x = (const float*)d_in[0];
  const int*   k = (const int*)d_in[1];
  float*     out = (float*)d_out;
  const int rows = in_sizes[0] / ROW_LEN;   // 4096
  topk_relu_scatter_kernel<<<rows, BLOCK, 0, stream>>>(x, k, out);
}